// TransR_63608465654044
// MI455X (gfx1250) — compile-verified
//
#include <hip/hip_runtime.h>
#include <hip/hip_bf16.h>

typedef float v2f __attribute__((ext_vector_type(2)));
typedef float v8f __attribute__((ext_vector_type(8)));
typedef int   v4i __attribute__((ext_vector_type(4)));

#define BATCH     65536
#define NREL      500
#define HID       128
#define RDIM      64
#define TILE      16
#define GELEMS    64                         // elements per group (4 WMMA tiles)
#define MAXGROUPS (NREL + BATCH / GELEMS)    // 1524 worst-case groups
#define EPSV      1e-12f
#define PSTRIDE   72                         // padded LDS row stride for proj (bank-conflict-free halves)
#define SSTRIDE   65                         // padded LDS row stride for staged outputs

// ---- workspace layout (in int32 units) ----
#define WS_COUNTS   0        // [500]
#define WS_OFFSETS  512      // [501]
#define WS_CURSOR   1024     // [500]
#define WS_NGROUPS  1536     // [1]
#define WS_PERM     2048     // [65536]
#define WS_GREL     67584    // [1536]
#define WS_GSTART   69120    // [1536]
#define WS_GCNT     70656    // [1536]
// total 72192 ints = ~288 KB

// ---- gfx1250 async global->LDS path (guarded; sync fallback keeps compile clean) ----
#if defined(__has_builtin)
# if __has_builtin(__builtin_amdgcn_global_load_async_to_lds_b128)
#  define HAS_ASYNC_LDS 1
# endif
#endif
#ifndef HAS_ASYNC_LDS
# define HAS_ASYNC_LDS 0
#endif

#define AS_GLOBAL __attribute__((address_space(1)))
#define AS_LDS    __attribute__((address_space(3)))

__device__ __forceinline__ void wait_asynccnt0() {
#if defined(__has_builtin) && __has_builtin(__builtin_amdgcn_s_wait_asynccnt)
  __builtin_amdgcn_s_wait_asynccnt(0);
#else
  asm volatile("s_wait_asynccnt 0x0" ::: "memory");
#endif
}

__global__ __launch_bounds__(256) void tr_init(int* ws) {
  int t = blockIdx.x * blockDim.x + threadIdx.x;
  if (t < NREL) ws[WS_COUNTS + t] = 0;
}

__global__ __launch_bounds__(256) void tr_count(const int* __restrict__ rel, int* ws) {
  int b = blockIdx.x * blockDim.x + threadIdx.x;
  if (b < BATCH) atomicAdd(&ws[WS_COUNTS + rel[b]], 1);
}

__global__ __launch_bounds__(512) void tr_scan(int* ws) {
  __shared__ int sc[512];
  __shared__ int sg[512];
  int t = threadIdx.x;
  int c  = (t < NREL) ? ws[WS_COUNTS + t] : 0;
  int ng = (c + GELEMS - 1) >> 6;
  sc[t] = c;
  sg[t] = ng;
  __syncthreads();
  for (int off = 1; off < 512; off <<= 1) {
    int vc = (t >= off) ? sc[t - off] : 0;
    int vg = (t >= off) ? sg[t - off] : 0;
    __syncthreads();
    sc[t] += vc;
    sg[t] += vg;
    __syncthreads();
  }
  int excC = sc[t] - c;   // exclusive element offset
  int excG = sg[t] - ng;  // exclusive group offset
  if (t < NREL) {
    ws[WS_OFFSETS + t] = excC;
    ws[WS_CURSOR  + t] = excC;
    for (int i = 0; i < ng; ++i) {
      int idx = excG + i;
      ws[WS_GREL   + idx] = t;
      ws[WS_GSTART + idx] = excC + GELEMS * i;
      int rem = c - GELEMS * i;
      ws[WS_GCNT   + idx] = rem < GELEMS ? rem : GELEMS;
    }
  }
  if (t == NREL - 1) {
    ws[WS_OFFSETS + NREL] = sc[t];
    ws[WS_NGROUPS] = sg[t];
  }
}

__global__ __launch_bounds__(256) void tr_scatter(const int* __restrict__ rel, int* ws) {
  int b = blockIdx.x * blockDim.x + threadIdx.x;
  if (b < BATCH) {
    int pos = atomicAdd(&ws[WS_CURSOR + rel[b]], 1);
    ws[WS_PERM + pos] = b;
  }
}

// One 128-thread block (4 wave32) per group of <=64 elements sharing one relation.
// Proj matrix (32KB) staged in LDS once (async b128 copies), each wave computes a
// 16x128 @ 128x64 f32 GEMM tile for src and dst rows with V_WMMA_F32_16X16X4_F32.
__global__ __launch_bounds__(128) void tr_wmma(
    const int*   __restrict__ src,
    const int*   __restrict__ dst,
    const float* __restrict__ ent,     // [NUM_NODES, 128]
    const float* __restrict__ relemb,  // [NREL, 64]
    const float* __restrict__ proj,    // [NREL, 128, 64]
    const int*   __restrict__ ws,
    float*       __restrict__ out) {
  int g = blockIdx.x;
  if (g >= ws[WS_NGROUPS]) return;          // uniform exit: EXEC stays all-ones

  const int t    = threadIdx.x;
  const int w    = t >> 5;                  // wave id (0..3) -> 16-row tile
  const int l    = t & 31;
  const int m    = l & 15;                  // row / column-within-ntile index
  const int half = l >> 4;                  // A/B K-pair selector

  const int r     = ws[WS_GREL   + g];
  const int start = ws[WS_GSTART + g];
  const int cnt   = ws[WS_GCNT   + g];

  __shared__ float lsP[HID * PSTRIDE];      // proj tile, padded rows  (~36 KB)
  __shared__ float lsS[4][TILE * SSTRIDE];  // staged src projections  (~16.6 KB)
  __shared__ float lsD[4][TILE * SSTRIDE];  // staged dst projections  (~16.6 KB)
  __shared__ float invb[4][32];             // per-row inverse norms

  // ---- stage proj[r] (128x64 f32) into LDS, padded to PSTRIDE per row ----
  const float* __restrict__ pbase = proj + (long)r * (HID * RDIM);
  #pragma unroll
  for (int i = 0; i < 16; ++i) {
    int chunk = t + 128 * i;                // 2048 float4 chunks
    int row   = chunk >> 4;
    int col   = (chunk & 15) << 2;
    const float* gp = pbase + row * RDIM + col;
    float*       lp = &lsP[row * PSTRIDE + col];
#if HAS_ASYNC_LDS
    __builtin_amdgcn_global_load_async_to_lds_b128(
        (AS_GLOBAL v4i*)(AS_GLOBAL void*)(AS_GLOBAL const void*)gp,
        (AS_LDS    v4i*)(AS_LDS    void*)lp, 0, 0);
#else
    *(float4*)lp = *(const float4*)gp;
#endif
  }
#if HAS_ASYNC_LDS
  wait_asynccnt0();
#endif
  __syncthreads();                          // proj visible to all 4 waves

  const int gm = TILE * w + m;              // this lane's group-row
  if (TILE * w < cnt) {                     // wave-uniform: EXEC stays all-ones
    const int elem = ws[WS_PERM + start + (gm < cnt ? gm : 0)];
    const float* __restrict__ srow = ent + (long)src[elem] * HID;
    const float* __restrict__ drow = ent + (long)dst[elem] * HID;

    v8f accS0 = {}, accS1 = {}, accS2 = {}, accS3 = {};
    v8f accD0 = {}, accD1 = {}, accD2 = {}, accD3 = {};

    #pragma unroll 4
    for (int kc = 0; kc < HID / 4; ++kc) {
      const int k = kc * 4 + half * 2;
      // A layout (f32 16x4): lanes 0-15 -> K=4kc..4kc+1, lanes 16-31 -> 4kc+2..4kc+3
      v2f aS = *(const v2f*)(srow + k);
      v2f aD = *(const v2f*)(drow + k);
      // B fragments from LDS: rows K,K+1 (lanes 0-15) / K+2,K+3 (lanes 16-31), col n
      const float* bp = &lsP[k * PSTRIDE + m];
      {
        v2f b = { bp[0],  bp[PSTRIDE + 0]  };
        accS0 = __builtin_amdgcn_wmma_f32_16x16x4_f32(false, aS, false, b, (short)0, accS0, false, false);
        accD0 = __builtin_amdgcn_wmma_f32_16x16x4_f32(false, aD, false, b, (short)0, accD0, false, false);
      }
      {
        v2f b = { bp[16], bp[PSTRIDE + 16] };
        accS1 = __builtin_amdgcn_wmma_f32_16x16x4_f32(false, aS, false, b, (short)0, accS1, false, false);
        accD1 = __builtin_amdgcn_wmma_f32_16x16x4_f32(false, aD, false, b, (short)0, accD1, false, false);
      }
      {
        v2f b = { bp[32], bp[PSTRIDE + 32] };
        accS2 = __builtin_amdgcn_wmma_f32_16x16x4_f32(false, aS, false, b, (short)0, accS2, false, false);
        accD2 = __builtin_amdgcn_wmma_f32_16x16x4_f32(false, aD, false, b, (short)0, accD2, false, false);
      }
      {
        v2f b = { bp[48], bp[PSTRIDE + 48] };
        accS3 = __builtin_amdgcn_wmma_f32_16x16x4_f32(false, aS, false, b, (short)0, accS3, false, false);
        accD3 = __builtin_amdgcn_wmma_f32_16x16x4_f32(false, aD, false, b, (short)0, accD3, false, false);
      }
    }

    // ---- spill C/D tiles to LDS row-major (padded) ----
    // C/D layout: VGPR j of lane l holds M = j + 8*half, N = nt*16 + m.
    const v8f* aS4[4] = { &accS0, &accS1, &accS2, &accS3 };
    const v8f* aD4[4] = { &accD0, &accD1, &accD2, &accD3 };
    #pragma unroll
    for (int nt = 0; nt < 4; ++nt) {
      #pragma unroll
      for (int j = 0; j < 8; ++j) {
        int row = j + 8 * half;
        int col = nt * 16 + m;
        lsS[w][row * SSTRIDE + col] = (*aS4[nt])[j];
        lsD[w][row * SSTRIDE + col] = (*aD4[nt])[j];
      }
    }

    // ---- pass 1: all 32 lanes compute one row's sum of squares ----
    // lanes 0-15: src row m ; lanes 16-31: dst row m
    {
      const float* rowp = half ? &lsD[w][m * SSTRIDE] : &lsS[w][m * SSTRIDE];
      float ss = 0.f;
      #pragma unroll 8
      for (int i = 0; i < RDIM; ++i) { float v = rowp[i]; ss += v * v; }
      invb[w][l] = 1.0f / fmaxf(sqrtf(ss), EPSV);
    }

    // ---- pass 2: lanes 0..cnt-1 finish their element ----
    if (half == 0 && gm < cnt) {
      const int e2 = ws[WS_PERM + start + gm];
      const float* __restrict__ re = relemb + (long)r * RDIM;
      const float* sRow = &lsS[w][m * SSTRIDE];
      const float* dRow = &lsD[w][m * SSTRIDE];
      const float invS = invb[w][m];
      const float invD = invb[w][16 + m];
      float acc = 0.f;
      #pragma unroll 8
      for (int i = 0; i < RDIM; ++i) {
        float diff = sRow[i] * invS + re[i] - dRow[i] * invD;
        acc += diff * diff;
      }
      out[e2] = sqrtf(acc);
    }
  }
}

extern "C" void kernel_launch(void* const* d_in, const int* in_sizes, int n_in,
                              void* d_out, int out_size, void* d_ws, size_t ws_size,
                              hipStream_t stream) {
  (void)in_sizes; (void)n_in; (void)out_size; (void)ws_size;
  const int*   src    = (const int*)d_in[0];
  const int*   rel    = (const int*)d_in[1];
  const int*   dst    = (const int*)d_in[2];
  const float* ent    = (const float*)d_in[3];
  const float* relemb = (const float*)d_in[4];
  const float* proj   = (const float*)d_in[5];
  float*       out    = (float*)d_out;
  int*         ws     = (int*)d_ws;

  tr_init   <<<(NREL  + 255) / 256, 256, 0, stream>>>(ws);
  tr_count  <<<(BATCH + 255) / 256, 256, 0, stream>>>(rel, ws);
  tr_scan   <<<1, 512, 0, stream>>>(ws);
  tr_scatter<<<(BATCH + 255) / 256, 256, 0, stream>>>(rel, ws);
  tr_wmma   <<<MAXGROUPS, 128, 0, stream>>>(src, dst, ent, relemb, proj, ws, out);
}